// SpearmanLoss_86535001080023
// MI455X (gfx1250) — compile-verified
//
#include <hip/hip_runtime.h>
#include <hip/hip_bf16.h>
#include <math.h>

// Spearman soft-rank loss, n = 4096, single-WGP / single-workgroup design.
// All state lives in LDS; PAV replaces the reference's O(n^2) isotonic formula.

#define N      4096
#define BLOCK  1024
#define NWAVE  (BLOCK / 32)     // 32 wave32 waves

typedef __attribute__((ext_vector_type(2))) float v2f;
typedef __attribute__((ext_vector_type(8))) float v8f;

#if defined(__has_builtin)
#if __has_builtin(__builtin_amdgcn_global_load_async_to_lds_b128) && \
    __has_builtin(__builtin_amdgcn_s_wait_asynccnt)
#define USE_ASYNC_LDS 1
#endif
#endif
#ifndef USE_ASYNC_LDS
#define USE_ASYNC_LDS 0
#endif

#if USE_ASYNC_LDS
// Builtin expects pointers to a 16-byte int vector (per hipcc diagnostic).
typedef int v4i_async __attribute__((vector_size(4 * sizeof(int))));
#endif

__device__ __forceinline__ float wave_reduce_sum(float v) {
#pragma unroll
  for (int o = 16; o > 0; o >>= 1) v += __shfl_xor(v, o, 32);
  return v;
}

// Reduction of 1024 per-thread register partials (shuffle + LDS tree).
__device__ float block_reduce_sum(float v, float* scratch, int tid) {
  v = wave_reduce_sum(v);
  if ((tid & 31) == 0) scratch[tid >> 5] = v;
  __syncthreads();
  if (tid < 32) {
    float w = scratch[tid];
    w = wave_reduce_sum(w);
    if (tid == 0) scratch[32] = w;
  }
  __syncthreads();
  float r = scratch[32];
  __syncthreads();
  return r;
}

// Sum of a 4096-element LDS array via fp32 WMMA "ones" matmul:
// D = A(16x4) * ones(4x16) + C. Each WMMA folds 64 f32 into the accumulator.
// Every column of D is identical -> summing all 8 accum VGPRs over all 32
// lanes yields 16x the true total; divide by 16, then tree-reduce.
// Must be called by all 1024 threads, fully converged (WMMA needs EXEC==~0).
__device__ float wmma_reduce_4096(const float* arr, float* scratch, int tid) {
  const int lane = tid & 31;
  const int wave = tid >> 5;
  v8f acc = {0.f, 0.f, 0.f, 0.f, 0.f, 0.f, 0.f, 0.f};
  v2f ones;
  ones[0] = 1.0f;
  ones[1] = 1.0f;
  const int base = wave * (N / NWAVE);             // 128 elements per wave
#pragma unroll
  for (int it = 0; it < (N / NWAVE) / 64; ++it) {  // 2 WMMAs per wave
    v2f a;
    a[0] = arr[base + it * 64 + 2 * lane];
    a[1] = arr[base + it * 64 + 2 * lane + 1];
    // 8 args: (neg_a, A, neg_b, B, c_mod, C, reuse_a, reuse_b)
    acc = __builtin_amdgcn_wmma_f32_16x16x4_f32(
        /*neg_a=*/false, a, /*neg_b=*/false, ones,
        /*c_mod=*/(short)0, acc, /*reuse_a=*/false, /*reuse_b=*/false);
  }
  float v = 0.0f;
#pragma unroll
  for (int r = 0; r < 8; ++r) v += acc[r];
  v *= 0.0625f;  // each of 16 columns holds a full copy of the row sums
  v = wave_reduce_sum(v);
  if (lane == 0) scratch[wave] = v;
  __syncthreads();
  if (tid < 32) {
    float w = scratch[tid];
    w = wave_reduce_sum(w);
    if (tid == 0) scratch[32] = w;
  }
  __syncthreads();
  float r = scratch[32];
  __syncthreads();
  return r;
}

__global__ void __launch_bounds__(BLOCK, 1)
spearman_loss_kernel(const float* __restrict__ pred,
                     const float* __restrict__ target,
                     float* __restrict__ out) {
  __shared__ __align__(16) float s_key[N];  // theta sorted desc; later rank scatter (16 KB)
  __shared__ __align__(16) float s_tmp[N];  // target stage / reduce scratch / PAV means (16 KB)
  __shared__ unsigned short s_idx[N];       // argsort payload (8 KB)
  __shared__ unsigned short s_bstart[N];    // PAV block start positions (8 KB)
  __shared__ float s_red[33];
  __shared__ int s_nb;

  const int tid = threadIdx.x;

  // ---- Stage pred -> s_key, target -> s_tmp -----------------------------
#if USE_ASYNC_LDS
  {
    // One b128 per lane per array: 1024 lanes x 16 B = 16 KB, ASYNCcnt-tracked,
    // no VGPR round-trip (gfx1250 GLOBAL_LOAD_ASYNC_TO_LDS_B128).
    __builtin_amdgcn_global_load_async_to_lds_b128(
        (v4i_async*)(pred + 4 * tid),
        (v4i_async*)(&s_key[4 * tid]), 0, 0);
    __builtin_amdgcn_global_load_async_to_lds_b128(
        (v4i_async*)(target + 4 * tid),
        (v4i_async*)(&s_tmp[4 * tid]), 0, 0);
    __builtin_amdgcn_s_wait_asynccnt(0);
  }
  __syncthreads();
#else
  __builtin_prefetch(pred + tid * 4, 0, 0);  // global_prefetch_b8
  __builtin_prefetch(target + tid * 4, 0, 0);
#pragma unroll
  for (int k = 0; k < N / BLOCK; ++k) {
    int j = tid + k * BLOCK;
    s_key[j] = pred[j];  // theta = pred / reg_strength (=1)
    s_tmp[j] = target[j];
  }
  __syncthreads();
#endif

  // ---- Phase A: MSE / target statistics (WMMA ones-matmul reductions) ----
  const float St = wmma_reduce_4096(s_tmp, s_red, tid);  // sum(t)

  float treg[N / BLOCK];
#pragma unroll
  for (int k = 0; k < N / BLOCK; ++k) {
    int j = tid + k * BLOCK;
    treg[k] = s_tmp[j];
    s_idx[j] = (unsigned short)j;
    float d = s_key[j] - treg[k];
    s_tmp[j] = d * d;                       // own-index rewrite, no race
  }
  __syncthreads();
  const float mse = wmma_reduce_4096(s_tmp, s_red, tid) * (1.0f / (float)N);

#pragma unroll
  for (int k = 0; k < N / BLOCK; ++k) {
    int j = tid + k * BLOCK;
    s_tmp[j] = treg[k] * treg[k];
  }
  __syncthreads();
  const float Stt = wmma_reduce_4096(s_tmp, s_red, tid);  // sum(t^2)

  // ---- Phase B: bitonic sort (descending) of (key, idx) in LDS -----------
  for (int k = 2; k <= N; k <<= 1) {
    for (int j = k >> 1; j > 0; j >>= 1) {
      __syncthreads();
      for (int i = tid; i < N; i += BLOCK) {
        int ixj = i ^ j;
        if (ixj > i) {
          float a = s_key[i];
          float b = s_key[ixj];
          // descending overall: in (i&k)==0 regions larger element goes first
          bool doSwap = ((i & k) == 0) ? (a < b) : (a > b);
          if (doSwap) {
            s_key[i] = b;
            s_key[ixj] = a;
            unsigned short ti = s_idx[i];
            s_idx[i] = s_idx[ixj];
            s_idx[ixj] = ti;
          }
        }
      }
    }
  }
  __syncthreads();

  // ---- Phase C: PAV on z[j] = s[j] - (N - j), non-increasing fit ---------
  // Exact equivalent of the reference's O(n^2) min-max isotonic formula.
  if (tid == 0) {
    int nb = 0;
    for (int j = 0; j < N; ++j) {
      float m = s_key[j] - (float)(N - j);
      int bs = j;
      while (nb > 0 && s_tmp[nb - 1] < m) {  // violation: prev mean < cur mean
        int ps = (int)s_bstart[nb - 1];
        float pc = (float)(bs - ps);
        float cc = (float)(j - bs + 1);
        m = (s_tmp[nb - 1] * pc + m * cc) / (pc + cc);
        bs = ps;
        --nb;
      }
      s_tmp[nb] = m;                          // block mean
      s_bstart[nb] = (unsigned short)bs;      // block start
      ++nb;
    }
    s_nb = nb;
  }
  __syncthreads();

  // ---- Phase D: ranks; scatter to original order; statistics -------------
  const int nb = s_nb;
  float rv[N / BLOCK];
#pragma unroll
  for (int k = 0; k < N / BLOCK; ++k) {
    int j = tid + k * BLOCK;
    // find last block with start <= j
    int lo = 0, hi = nb - 1;
    while (lo < hi) {
      int mid = (lo + hi + 1) >> 1;
      if ((int)s_bstart[mid] <= j) lo = mid;
      else hi = mid - 1;
    }
    rv[k] = (s_key[j] - s_tmp[lo]) * (1.0f / (float)N);  // soft rank / n
  }
  __syncthreads();
  // permutation scatter: rank back to original element order (s_key is dead)
#pragma unroll
  for (int k = 0; k < N / BLOCK; ++k) {
    int j = tid + k * BLOCK;
    s_key[(int)s_idx[j]] = rv[k];
  }
  __syncthreads();

  float pr = 0.f, prr = 0.f, prt = 0.f;
#pragma unroll
  for (int k = 0; k < N / BLOCK; ++k) {
    int j = tid + k * BLOCK;
    float r = s_key[j];
    float t = target[j];  // coalesced, L2-hot re-read (no random gather)
    pr  += r;
    prr += r * r;
    prt += r * t;
  }
  const float Sr  = block_reduce_sum(pr,  s_red, tid);
  const float Srr = block_reduce_sum(prr, s_red, tid);
  const float Srt = block_reduce_sum(prt, s_red, tid);

  // ---- Phase E: assemble loss -------------------------------------------
  if (tid == 0) {
    float n = (float)N;
    float rmean = Sr / n;
    float tmean = St / n;
    float cov  = Srt - n * rmean * tmean;    // sum(rn * tn), un-normalized
    float rden = sqrtf(Srr - n * rmean * rmean);
    float tden = sqrtf(Stt - n * tmean * tmean);
    out[0] = -0.1f * (cov / (rden * tden)) + mse;
  }
}

extern "C" void kernel_launch(void* const* d_in, const int* in_sizes, int n_in,
                              void* d_out, int out_size, void* d_ws, size_t ws_size,
                              hipStream_t stream) {
  const float* pred   = (const float*)d_in[0];
  const float* target = (const float*)d_in[1];
  float* out = (float*)d_out;
  (void)in_sizes; (void)n_in; (void)out_size; (void)d_ws; (void)ws_size;
  spearman_loss_kernel<<<1, BLOCK, 0, stream>>>(pred, target, out);
}